// FLFQuantizer_88467736363508
// MI455X (gfx1250) — compile-verified
//
#include <hip/hip_runtime.h>

typedef __attribute__((ext_vector_type(2))) float v2f;
typedef __attribute__((ext_vector_type(8))) float v8f;

#define TOKENS 2048   // 8 * 256
#define DIMW   128    // embedding dim
#define DC     16     // codebook dims (log2 of 65536)

// One wave (32 lanes) handles one 16-token tile.
// GEMM1: z = x_tile[16x128] @ W_in[128x16] + b_in          (32x v_wmma_f32_16x16x4_f32)
// sign(z) -> ballot masks -> code index (bit-reversed 16-bit sign mask)
// GEMM2: out = q[16x16] @ W_out[16x128] + b_out            (32x v_wmma_f32_16x16x4_f32)
__global__ __launch_bounds__(32) void flf_quantizer_wmma(
    const float* __restrict__ x,     // [2048,128]
    const float* __restrict__ Win,   // [128,16]
    const float* __restrict__ bin,   // [16]
    const float* __restrict__ Wout,  // [16,128]
    const float* __restrict__ bout,  // [128]
    float* __restrict__ out,         // [2048,128]
    float* __restrict__ idx_out)     // [2048]
{
    const int lane = threadIdx.x;        // 0..31
    const int nl   = lane & 15;          // N (or M) position within half-wave
    const int hi   = lane >> 4;          // which half of the wave
    const int kb   = 2 * hi;             // K base inside each K=4 chunk (A/B f32 layout)
    const int tile = blockIdx.x;         // 0..127
    const int row0 = tile * 16;

    // ---- Stage W_in as B-matrix fragments (tile-invariant): 32 chunks of K=4 ----
    // B layout (4x16 f32): lane holds N=nl, K = 4kk + kb + {0,1}
    v2f wb1[32];
#pragma unroll
    for (int kk = 0; kk < 32; ++kk) {
        const int k0 = 4 * kk + kb;
        wb1[kk].x = Win[(k0 + 0) * DC + nl];
        wb1[kk].y = Win[(k0 + 1) * DC + nl];
    }

    // ---- GEMM1: z[token 16 x dim 16] ; C initialized with b_in (N = dim = nl) ----
    v8f z;
    {
        const float biv = bin[nl];
#pragma unroll
        for (int r = 0; r < 8; ++r) z[r] = biv;
    }
#pragma unroll
    for (int kk = 0; kk < 32; ++kk) {
        const int k0 = 4 * kk + kb;
        // A layout (16x4 f32): lane holds M=nl (token), K = k0 + {0,1} -> contiguous float2
        v2f a;
        a.x = x[(row0 + nl) * DIMW + k0 + 0];
        a.y = x[(row0 + nl) * DIMW + k0 + 1];
        z = __builtin_amdgcn_wmma_f32_16x16x4_f32(false, a, false, wb1[kk],
                                                  (short)0, z, false, false);
    }

    // ---- sign(z) via wave ballots (results are wave-uniform) ----
    // z vgpr r, lane l = z[token r + 8*(l>>4)][dim l&15]
    // => ball[r] bit n      = (z[token r  ][dim n] > 0)
    //    ball[r] bit 16+n   = (z[token r+8][dim n] > 0)
    unsigned ball[8];
#pragma unroll
    for (int r = 0; r < 8; ++r)
        ball[r] = __builtin_amdgcn_ballot_w32(z[r] > 0.0f);

    // 16 sign bits (dims 0..15) of this lane's token m = nl
    const unsigned bitsrc = (ball[nl & 7] >> (16 * (nl >> 3))) & 0xFFFFu;

    // ---- quantized A-matrix fragments: q[m][k] = bit ? +1 : -1 ----
    v2f a2[4];
#pragma unroll
    for (int kk = 0; kk < 4; ++kk) {
        const int k0 = 4 * kk + kb;
        a2[kk].x = ((bitsrc >> (k0 + 0)) & 1u) ? 1.0f : -1.0f;
        a2[kk].y = ((bitsrc >> (k0 + 1)) & 1u) ? 1.0f : -1.0f;
    }

    // ---- GEMM2: out_tile[16 x 128] = q @ W_out + b_out, 8 N-tiles of 16 ----
#pragma unroll
    for (int j = 0; j < 8; ++j) {
        v8f acc;
        {
            const float bo = bout[16 * j + nl];
#pragma unroll
            for (int r = 0; r < 8; ++r) acc[r] = bo;
        }
#pragma unroll
        for (int kk = 0; kk < 4; ++kk) {
            const int k0 = 4 * kk + kb;
            v2f b;
            b.x = Wout[(k0 + 0) * DIMW + 16 * j + nl];
            b.y = Wout[(k0 + 1) * DIMW + 16 * j + nl];
            acc = __builtin_amdgcn_wmma_f32_16x16x4_f32(false, a2[kk], false, b,
                                                        (short)0, acc, false, false);
        }
        // D layout: vgpr r, lane l -> row m = r + 8*hi, col = 16j + nl (contiguous in lanes)
#pragma unroll
        for (int r = 0; r < 8; ++r) {
            const int m = r + 8 * hi;
            out[(row0 + m) * DIMW + 16 * j + nl] = acc[r];
        }
    }

    // ---- encoding indices: codebook is MSB-first, so idx = bit-reverse16(sign mask) ----
    if (lane < 16) {
        const unsigned msk = (ball[lane & 7] >> (16 * (lane >> 3))) & 0xFFFFu;
        const unsigned idx = __brev(msk) >> 16;   // reverse 16-bit field
        idx_out[row0 + lane] = (float)idx;
    }
}

extern "C" void kernel_launch(void* const* d_in, const int* in_sizes, int n_in,
                              void* d_out, int out_size, void* d_ws, size_t ws_size,
                              hipStream_t stream) {
    (void)in_sizes; (void)n_in; (void)out_size; (void)d_ws; (void)ws_size;
    const float* x    = (const float*)d_in[0];
    const float* Win  = (const float*)d_in[1];
    const float* bin  = (const float*)d_in[2];
    const float* Wout = (const float*)d_in[3];
    const float* bout = (const float*)d_in[4];
    float* out = (float*)d_out;                    // [2048*128] projected outputs
    float* idx = out + (size_t)TOKENS * DIMW;      // [2048] encoding indices (flat tuple)

    flf_quantizer_wmma<<<TOKENS / 16, 32, 0, stream>>>(x, Win, bin, Wout, bout, out, idx);
}